// SelectiveGroupModel_57595511439613
// MI455X (gfx1250) — compile-verified
//
#include <hip/hip_runtime.h>

#define D_DIM 4096
#define G_DIM 32
#define K_SEL 128
#define H_DIM 16
#define B_DIM 8192
#define SLOTS 144   // 128 + worst-case pad (4 chunks * 3) rounded up
#define CHUNK 1024
#define XPAD 1028   // LDS row stride in floats (16B aligned, bank-spread)

typedef __attribute__((ext_vector_type(2))) float v2f;
typedef __attribute__((ext_vector_type(8))) float v8f;

// ---------------- workspace layout (bytes) ----------------
// ssq      : 0       .. 16384   (4096 f32)
// corr_acc : 16384   .. 16512   (32 f32)
// poff     : 16512   .. 17152   (32*5 int)
// gidx     : 17152   .. 35584   (32*144 int)
// meta     : 35584   .. 72448   (32*144 int2)
// W1selT   : 72448   .. 367360  (32*16*144 f32, [g][h][slot])

// -------- Kernel 1: per-column sum of squares over all B rows --------
__global__ void k_ssq(const float* __restrict__ x, float* __restrict__ ssq) {
    int t = threadIdx.x;
    int r0 = blockIdx.x * 128;
    float acc[16];
#pragma unroll
    for (int i = 0; i < 16; ++i) acc[i] = 0.f;
    for (int r = 0; r < 128; ++r) {
        const float* row = x + (size_t)(r0 + r) * D_DIM;
#pragma unroll
        for (int i = 0; i < 16; ++i) {
            float v = row[t + 256 * i];
            acc[i] += v * v;
        }
    }
#pragma unroll
    for (int i = 0; i < 16; ++i) atomicAdd(&ssq[t + 256 * i], acc[i]);
}

// -------- Kernel 2: per-group top-K radix select + chunked compaction --------
__global__ void k_topk(const float* __restrict__ u, const float* __restrict__ gl,
                       const float* __restrict__ W1,
                       int* __restrict__ gidx, int* __restrict__ poff,
                       float* __restrict__ W1selT) {
    __shared__ unsigned key[D_DIM];
    __shared__ unsigned hist[256];
    __shared__ int tsum[256];
    __shared__ int slist[K_SEL];
    __shared__ int cnt[4];
    __shared__ int lo4[4];
    __shared__ int pf[5];
    __shared__ unsigned sh_prefix;
    __shared__ int sh_rem;

    int g = blockIdx.x;
    int t = threadIdx.x;

    // keys: monotone uint mapping of z = logits + gumbel
    for (int d = t; d < D_DIM; d += 256) {
        float uu = u[(size_t)g * D_DIM + d];
        float z  = gl[(size_t)g * D_DIM + d] - logf(-logf(uu));
        unsigned b = __float_as_uint(z);
        key[d] = (b & 0x80000000u) ? ~b : (b | 0x80000000u);
    }
    if (t == 0) { sh_prefix = 0u; sh_rem = K_SEL; }
    __syncthreads();

    // 4-pass MSB radix select: find threshold key T and tie count
    for (int p = 3; p >= 0; --p) {
        hist[t] = 0u;
        __syncthreads();
        unsigned prefix = sh_prefix;
        int shift = 8 * p;
        for (int j = 0; j < 16; ++j) {
            unsigned k = key[t * 16 + j];
            bool match = (p == 3) || ((k >> (shift + 8)) == prefix);
            if (match) atomicAdd(&hist[(k >> shift) & 255u], 1u);
        }
        __syncthreads();
        if (t == 0) {
            int rem = sh_rem;
            unsigned acc = 0;
            for (int b = 255; b >= 0; --b) {
                unsigned h = hist[b];
                acc += h;
                if ((int)acc >= rem) {
                    sh_prefix = (sh_prefix << 8) | (unsigned)b;
                    sh_rem = rem - (int)(acc - h);
                    break;
                }
            }
        }
        __syncthreads();
    }
    unsigned T = sh_prefix;
    int rem = sh_rem;

    // tie-rank exclusive scan (thread owns 16 consecutive d's)
    int tloc = 0;
    for (int j = 0; j < 16; ++j) if (key[t * 16 + j] == T) tloc++;
    tsum[t] = tloc;
    __syncthreads();
    if (t == 0) { int run = 0; for (int i = 0; i < 256; ++i) { int v = tsum[i]; tsum[i] = run; run += v; } }
    __syncthreads();
    int tiebase = tsum[t];

    // selection count per thread
    int sloc = 0;
    {
        int tb = tiebase;
        for (int j = 0; j < 16; ++j) {
            unsigned k = key[t * 16 + j];
            bool sel = (k > T) || (k == T && tb < rem);
            if (k == T) tb++;
            if (sel) sloc++;
        }
    }
    __syncthreads();
    tsum[t] = sloc;
    __syncthreads();
    if (t == 0) { int run = 0; for (int i = 0; i < 256; ++i) { int v = tsum[i]; tsum[i] = run; run += v; } }
    __syncthreads();

    // emit selected indices in ascending order
    {
        int pos = tsum[t];
        int tb = tiebase;
        for (int j = 0; j < 16; ++j) {
            unsigned k = key[t * 16 + j];
            bool sel = (k > T) || (k == T && tb < rem);
            if (k == T) tb++;
            if (sel) { slist[pos] = t * 16 + j; pos++; }
        }
    }
    if (t < 4) cnt[t] = 0;
    __syncthreads();
    if (t < K_SEL) atomicAdd((unsigned*)&cnt[slist[t] >> 10], 1u);
    __syncthreads();
    if (t == 0) {
        lo4[0] = 0; pf[0] = 0;
        int lorun = 0, porun = 0;
        for (int c = 0; c < 4; ++c) {
            if (c > 0) lo4[c] = lorun;
            porun += (cnt[c] + 3) & ~3;
            lorun += cnt[c];
            pf[c + 1] = porun;
        }
        for (int c = 0; c < 5; ++c) poff[g * 5 + c] = pf[c];
    }
    __syncthreads();
    for (int s = t; s < SLOTS; s += 256) gidx[g * SLOTS + s] = -1;
    __syncthreads();
    if (t < K_SEL) {
        int d = slist[t];
        int c = d >> 10;
        gidx[g * SLOTS + pf[c] + (t - lo4[c])] = d;
    }
    __syncthreads();
    // gather selected W1 rows, TRANSPOSED: W1selT[g][h][slot] (pads -> 0)
    for (int e = t; e < SLOTS * H_DIM; e += 256) {
        int h = e / SLOTS, slot = e - h * SLOTS;
        int gd = gidx[g * SLOTS + slot];
        W1selT[((size_t)g * H_DIM + h) * SLOTS + slot] =
            (gd >= 0) ? W1[((size_t)g * D_DIM + gd) * H_DIM + h] : 0.f;
    }
}

// -------- Kernel 3: pack {local column, inv-norm} per slot --------
__global__ void k_meta(const int* __restrict__ gidx, const float* __restrict__ ssq,
                       int2* __restrict__ meta) {
    int g = blockIdx.x;
    int s = threadIdx.x;
    if (s < SLOTS) {
        int gd = gidx[g * SLOTS + s];
        int2 m;
        if (gd >= 0) {
            float nv = fmaxf(sqrtf(ssq[gd]), 1e-12f);
            m.x = gd & (CHUNK - 1);
            m.y = __float_as_int(1.f / nv);
        } else {
            m.x = 0;
            m.y = 0;   // inv-norm 0 => pad contributes nothing to s
        }
        meta[g * SLOTS + s] = m;
    }
}

// -------- Kernel 4: fused gathered GEMM (WMMA f32) + row sums --------
__global__ void __launch_bounds__(256, 1)
k_main(const float* __restrict__ x, const float* __restrict__ W1selT,
       const int2* __restrict__ meta, const int* __restrict__ poff,
       const float* __restrict__ b1, const float* __restrict__ W2,
       const float* __restrict__ b2,
       float* __restrict__ out, float* __restrict__ corr_acc) {
    __shared__ float xs[16 * XPAD];                          // 65,792 B
    __shared__ __align__(16) int2 metas[G_DIM * SLOTS];      // 36,864 B

    int t = threadIdx.x;
    int lane = t & 31;
    int wave = t >> 5;
    int b0 = blockIdx.x * 16;
    int m = lane & 15;                 // A-matrix row / B-matrix col / C col
    int koff = (lane < 16) ? 0 : 2;    // K split across half-waves (16x4 f32 layout)
    int gbase = wave * 4;              // 8 waves * 4 groups = 32
    const float* xrow = xs + m * XPAD;

    for (int i = t; i < G_DIM * SLOTS; i += 256) metas[i] = meta[i];

    v8f acc[4];
    float sacc[4];
#pragma unroll
    for (int gi = 0; gi < 4; ++gi) {
        acc[gi] = (v8f){0.f, 0.f, 0.f, 0.f, 0.f, 0.f, 0.f, 0.f};
        sacc[gi] = 0.f;
    }

    for (int c = 0; c < 4; ++c) {
        __syncthreads();   // protect xs from previous iteration's readers
        // stage 16 rows x 1024 cols of x straight into LDS via CDNA5
        // async-to-LDS copies (ASYNCcnt-tracked, no VGPR roundtrip)
#pragma unroll
        for (int i = 0; i < 16; ++i) {
            unsigned lofs = (unsigned)(uintptr_t)&xs[i * XPAD + t * 4];
            const float* gp = x + (size_t)(b0 + i) * D_DIM + c * CHUNK + t * 4;
            asm volatile("global_load_async_to_lds_b128 %0, %1, off"
                         :: "v"(lofs), "v"(gp) : "memory");
        }
        asm volatile("s_wait_asynccnt 0x0" ::: "memory");
        __syncthreads();

#pragma unroll
        for (int gi = 0; gi < 4; ++gi) {
            int g = gbase + gi;
            // scalar (wave-uniform) trip count -> uniform loop, EXEC untouched
            int base = __builtin_amdgcn_readfirstlane(poff[g * 5 + c]);
            int nst  = (__builtin_amdgcn_readfirstlane(poff[g * 5 + c + 1]) - base) >> 2;
            if (nst <= 0) continue;

            const int2* mp = metas + g * SLOTS;
            int sI = base + koff;
            const float* wp = W1selT + ((size_t)g * H_DIM + m) * SLOTS + sI;

            // software pipeline: stage-0 prefetch (over-read by one stage is
            // in-bounds: max slot index = poff[c+1] + koff + 1 <= 143 < SLOTS)
            int4   mm = *(const int4*)(mp + sI);     // {lcol0, inv0, lcol1, inv1}
            float2 bb = *(const float2*)(wp);        // B fragment (2 consecutive slots)

            for (int it = 0; it < nst; ++it) {
                float a0 = xrow[mm.x];
                float a1 = xrow[mm.z];
                float inv0 = __int_as_float(mm.y);
                float inv1 = __int_as_float(mm.w);
                // prefetch next stage before the WMMA
                sI += 4; wp += 4;
                int4   mmn = *(const int4*)(mp + sI);
                float2 bbn = *(const float2*)(wp);

                sacc[gi] = fmaf(a0, inv0, fmaf(a1, inv1, sacc[gi]));
                v2f av = {a0, a1};
                v2f bv = {bb.x, bb.y};
                acc[gi] = __builtin_amdgcn_wmma_f32_16x16x4_f32(
                    false, av, false, bv, (short)0, acc[gi], false, false);
                mm = mmn; bb = bbn;
            }
        }
    }

    // epilogue: relu(y+b1)*W2 reduced over H via shfl, plus sum(s^2) per group
    int rbase = (lane < 16) ? 0 : 8;   // C layout: VGPR v -> M = v (+8 for upper half)
#pragma unroll
    for (int gi = 0; gi < 4; ++gi) {
        int g = gbase + gi;
        float b1v = b1[g * H_DIM + m];
        float w2v = W2[g * H_DIM + m];
        float b2v = b2[g];
#pragma unroll
        for (int v = 0; v < 8; ++v) {
            float p = fmaxf(acc[gi][v] + b1v, 0.f) * w2v;
            p += __shfl_xor(p, 1);
            p += __shfl_xor(p, 2);
            p += __shfl_xor(p, 4);
            p += __shfl_xor(p, 8);
            if (m == v) out[(size_t)(b0 + rbase + v) * G_DIM + g] = p + b2v;
        }
        float s = sacc[gi] + __shfl_xor(sacc[gi], 16);  // combine K%4 halves per row
        float s2 = s * s;
        s2 += __shfl_xor(s2, 1);
        s2 += __shfl_xor(s2, 2);
        s2 += __shfl_xor(s2, 4);
        s2 += __shfl_xor(s2, 8);
        if (lane == 0) atomicAdd(&corr_acc[g], s2);
    }
}

// -------- Kernel 5: closed-form correlations --------
__global__ void k_corr(const float* __restrict__ accum, float* __restrict__ corr_out) {
    int g = threadIdx.x;
    if (g < G_DIM) {
        float S = accum[g];
        float val = (S - (float)K_SEL) / (2.f * (float)B_DIM);
        corr_out[g] = val / ((float)(K_SEL * (K_SEL - 1)) * 0.5f + 1e-6f);
    }
}

extern "C" void kernel_launch(void* const* d_in, const int* in_sizes, int n_in,
                              void* d_out, int out_size, void* d_ws, size_t ws_size,
                              hipStream_t stream) {
    const float* x  = (const float*)d_in[0];
    const float* u  = (const float*)d_in[1];
    const float* gl = (const float*)d_in[2];
    const float* W1 = (const float*)d_in[3];
    const float* b1 = (const float*)d_in[4];
    const float* W2 = (const float*)d_in[5];
    const float* b2 = (const float*)d_in[6];
    float* out = (float*)d_out;
    float* corr_out = out + (size_t)B_DIM * G_DIM;

    char* ws = (char*)d_ws;
    float* ssq      = (float*)(ws + 0);
    float* corr_acc = (float*)(ws + 16384);
    int*   poff     = (int*)  (ws + 16512);
    int*   gidx     = (int*)  (ws + 17152);
    int2*  meta     = (int2*) (ws + 35584);
    float* W1selT   = (float*)(ws + 72448);

    hipMemsetAsync(ws, 0, 16512, stream);                 // ssq + corr_acc
    k_ssq <<<64,  256, 0, stream>>>(x, ssq);
    k_topk<<<G_DIM, 256, 0, stream>>>(u, gl, W1, gidx, poff, W1selT);
    k_meta<<<G_DIM, 160, 0, stream>>>(gidx, ssq, meta);
    k_main<<<B_DIM / 16, 256, 0, stream>>>(x, W1selT, meta, poff, b1, W2, b2, out, corr_acc);
    k_corr<<<1, 32, 0, stream>>>(corr_acc, corr_out);
}